// SpectralColorOrientLoss3D_9655086481799
// MI455X (gfx1250) — compile-verified
//
#include <hip/hip_runtime.h>

// SpectralColorOrientLoss3D — fused single-pass reduction for MI455X (gfx1250).
// Memory-bound: ~201 MB read, ideal ~8.6us @ 23.3 TB/s. Wave32 layout:
// one wave per contiguous 128-float row (32 lanes x float4), WMMA f32 16x16x4
// used as an exact fp32 32-lane reduction, fp64 atomics for cross-block sums.

typedef float v2f __attribute__((ext_vector_type(2)));
typedef float v8f __attribute__((ext_vector_type(8)));

#define B_ 2
#define S_ 128
#define R_ 128
#define C_ 128
#define NVOL (S_ * R_ * C_)   // 2097152

// Exact fp32 sum of all 32 lanes using one V_WMMA_F32_16X16X4_F32.
// A[m,0]=p_m (lanes 0-15), A[m,2]=p_{m+16} (lanes 16-31), A[m,1]=A[m,3]=0.
// B = ones(4x16)  =>  D[m,n] = p_m + p_{m+16}  (broadcast over n).
// D layout: lane l<16 holds D[0..7, l] in d[0..7]; lane l>=16 holds D[8..15, l-16].
// In-lane sum of d[0..7] gives half-sums split across lane halves; one xor-16
// shuffle + add yields the full 32-lane total in every lane.
__device__ __forceinline__ float wave_sum_wmma(float p) {
    v2f a;  a[0] = p;    a[1] = 0.0f;
    v2f bo; bo[0] = 1.0f; bo[1] = 1.0f;
    v8f c = {};
    v8f d = __builtin_amdgcn_wmma_f32_16x16x4_f32(
        /*neg_a=*/false, a, /*neg_b=*/false, bo,
        /*c_mod=*/(short)0, c, /*reuse_a=*/false, /*reuse_b=*/false);
    float h = ((d[0] + d[1]) + (d[2] + d[3])) + ((d[4] + d[5]) + (d[6] + d[7]));
    h += __shfl_xor(h, 16, 32);
    return h;
}

__device__ __forceinline__ int argmax3(float a, float b, float c) {
    int j = 0; float m = a;
    if (b > m) { j = 1; m = b; }
    if (c > m) { j = 2; }
    return j;   // strict '>' => first index wins ties, matching jnp.argmax
}

__global__ void zero_acc_kernel(double* acc) {
    if (threadIdx.x < 4) acc[threadIdx.x] = 0.0;
}

__global__ __launch_bounds__(256) void scol3d_fused_kernel(
        const float* __restrict__ L, const float* __restrict__ U,
        double* __restrict__ acc) {
    const int tid  = threadIdx.x;
    const int lane = tid & 31;
    const int wave = tid >> 5;
    const int row  = blockIdx.x * 8 + wave;   // 0 .. B*S*R-1 = 32767
    const int b    = row >> 14;               // / (S*R)
    const int sr   = row & 16383;
    const int s    = sr >> 7;
    const int r    = sr & 127;
    const int c    = lane << 2;               // 0,4,...,124
    const size_t p = (size_t)sr * C_ + (size_t)c;

    const float* Lb = L + (size_t)b * 3 * NVOL;
    const float* Ub = U + (size_t)b * 9 * NVOL;

    // ---- argmax over L channels (per voxel of the float4) ----
    const float4 l0 = *(const float4*)(Lb + p);
    const float4 l1 = *(const float4*)(Lb + (size_t)NVOL + p);
    const float4 l2 = *(const float4*)(Lb + (size_t)2 * NVOL + p);
    int jsel[4];
    jsel[0] = argmax3(l0.x, l1.x, l2.x);
    jsel[1] = argmax3(l0.y, l1.y, l2.y);
    jsel[2] = argmax3(l0.z, l1.z, l2.z);
    jsel[3] = argmax3(l0.w, l1.w, l2.w);

    const bool haveR = (r < R_ - 1);   // wave-uniform
    const bool haveS = (s < S_ - 1);   // wave-uniform

    float am[3][4];
#pragma unroll
    for (int k = 0; k < 3; ++k)
#pragma unroll
        for (int t = 0; t < 4; ++t) am[k][t] = 0.0f;

    float gc = 0.0f, gr = 0.0f, gs = 0.0f;

#pragma unroll
    for (int i = 0; i < 9; ++i) {
        const float* Ui = Ub + (size_t)i * NVOL;
        const float4 u  = *(const float4*)(Ui + p);

        // C-axis diffs: 3 in-register + 1 cross-lane (next lane's u.x)
        const float d0 = u.y - u.x, d1 = u.z - u.y, d2 = u.w - u.z;
        gc += d0 * d0 + d1 * d1 + d2 * d2;
        const float un = __shfl_down(u.x, 1, 32);
        const float d3 = un - u.w;
        gc += (lane < 31) ? d3 * d3 : 0.0f;   // cndmask, EXEC untouched

        // R-axis diffs (neighbor row: L2-resident, reuse distance 512B)
        if (haveR) {
            const float4 ur = *(const float4*)(Ui + p + C_);
            const float e0 = ur.x - u.x, e1 = ur.y - u.y;
            const float e2 = ur.z - u.z, e3 = ur.w - u.w;
            gr += e0 * e0 + e1 * e1 + e2 * e2 + e3 * e3;
        }
        // S-axis diffs (neighbor slice: L2-resident, reuse distance ~600KB)
        if (haveS) {
            const float4 us = *(const float4*)(Ui + p + (size_t)R_ * C_);
            const float f0 = us.x - u.x, f1 = us.y - u.y;
            const float f2 = us.z - u.z, f3 = us.w - u.w;
            gs += f0 * f0 + f1 * f1 + f2 * f2 + f3 * f3;
        }

        // Select |U[b, 3k+j, v]| for the CO term (register selects only)
        const int kk = i / 3, jj = i % 3;
        am[kk][0] = (jsel[0] == jj) ? fabsf(u.x) : am[kk][0];
        am[kk][1] = (jsel[1] == jj) ? fabsf(u.y) : am[kk][1];
        am[kk][2] = (jsel[2] == jj) ? fabsf(u.z) : am[kk][2];
        am[kk][3] = (jsel[3] == jj) ? fabsf(u.w) : am[kk][3];
    }

    // ---- CO per voxel ----
    float co_sum = 0.0f;
#pragma unroll
    for (int t = 0; t < 4; ++t) {
        const float e1 = am[0][t], e2 = am[1][t], e3 = am[2][t];
        float den = e1 * e1 + e2 * e2 + e3 * e3;
        den = (den < 1e-14f) ? 1e-14f : den;
        const float q0 = e1 - e2, q1 = e2 - e3, q2 = e3 - e1;
        const float num = q0 * q0 + q1 * q1 + q2 * q2;
        co_sum += sqrtf(0.5f * num / den);
    }

    // ---- reductions: WMMA wave sum (EXEC all-ones here), LDS across waves ----
    const float t0 = wave_sum_wmma(co_sum);
    const float t1 = wave_sum_wmma(gc);
    const float t2 = wave_sum_wmma(gr);
    const float t3 = wave_sum_wmma(gs);

    __shared__ float red[8][4];
    if (lane == 0) {
        red[wave][0] = t0; red[wave][1] = t1;
        red[wave][2] = t2; red[wave][3] = t3;
    }
    __syncthreads();
    if (tid < 4) {
        float sum = 0.0f;
#pragma unroll
        for (int w = 0; w < 8; ++w) sum += red[w][tid];
        atomicAdd(&acc[tid], (double)sum);   // global_atomic_add_f64
    }
}

__global__ void finalize_kernel(const double* __restrict__ acc,
                                float* __restrict__ out) {
    if (blockIdx.x == 0 && threadIdx.x == 0) {
        const double mean_co = acc[0] / (double)((size_t)B_ * NVOL);
        const double gC = acc[1] / (double)((size_t)B_ * S_ * R_ * (C_ - 1));
        const double gR = acc[2] / (double)((size_t)B_ * S_ * (R_ - 1) * C_);
        const double gS = acc[3] / (double)((size_t)B_ * (S_ - 1) * R_ * C_);
        out[0] = (float)(0.5 * (mean_co + gC + gR + gS));
    }
}

extern "C" void kernel_launch(void* const* d_in, const int* in_sizes, int n_in,
                              void* d_out, int out_size, void* d_ws, size_t ws_size,
                              hipStream_t stream) {
    const float* L = (const float*)d_in[0];   // (2,3,128,128,128) f32
    const float* U = (const float*)d_in[1];   // (2,9,128,128,128) f32
    double* acc = (double*)d_ws;              // 4 fp64 accumulators

    zero_acc_kernel<<<1, 32, 0, stream>>>(acc);
    // B*S*R = 32768 rows, 8 rows (8 waves) per 256-thread block
    scol3d_fused_kernel<<<4096, 256, 0, stream>>>(L, U, acc);
    finalize_kernel<<<1, 32, 0, stream>>>(acc, (float*)d_out);
}